// LowImpactLEEA_5652176962359
// MI455X (gfx1250) — compile-verified
//
#include <hip/hip_runtime.h>
#include <hip/hip_bf16.h>

// ============================================================================
// LowImpactLEEA — exact algebraic collapse.
//
//   softmax over axis=2 followed by sum over axis=2  =>  attn_agg == 1.
//   => out[b,n,d] = x[b,n,d] + sigmoid(gate) * (sum_s Wv[d,s] + bv[d])
//
// Kernel 1: one wave32 computes c[d] = sigmoid(gate)*(rowsum(Wv)[d] + bv[d])
//           using V_WMMA_F32_16X16X4_F32 with an all-ones B matrix (exact f32).
// Kernel 2: bandwidth-bound broadcast add, float4 (b128) stream of x -> out,
//           constant staged in LDS, software prefetch of the x stream.
// ============================================================================

typedef __attribute__((ext_vector_type(2))) float v2f;
typedef __attribute__((ext_vector_type(8))) float v8f;

#define D_MODEL 256
#define S_DIM   32

// ---- Kernel 1: constant vector via f32 WMMA (1 block, 1 wave) --------------
__global__ void __launch_bounds__(32)
leea_const_wmma_kernel(const float* __restrict__ Wv,   // [D_MODEL][S_DIM] row-major ('ds')
                       const float* __restrict__ bv,   // [D_MODEL]
                       const float* __restrict__ gate, // [1]
                       float* __restrict__ c_out)      // [D_MODEL]
{
    const int lane  = threadIdx.x;        // 0..31
    const float g   = gate[0];
    const float sig = 1.0f / (1.0f + __expf(-g));

    // B = all-ones 4x16: every column of D receives the 4-wide row sum of A.
    v2f bmat;
    bmat.x = 1.0f;
    bmat.y = 1.0f;

    // A-matrix (16x4 f32) VGPR layout: lanes 0-15 -> M=lane, K={0,1} in v0/v1;
    // lanes 16-31 -> M=lane-16, K={2,3}.
    const int mrow  = lane & 15;
    const int kbase = (lane >> 4) * 2;    // 0 or 2

    for (int t = 0; t < D_MODEL / 16; ++t) {          // 16 tiles of 16 rows
        v8f acc = {};
        const float* Arow = Wv + (size_t)(t * 16 + mrow) * S_DIM;
        #pragma unroll
        for (int kk = 0; kk < S_DIM / 4; ++kk) {      // 8 K-steps of 4
            v2f a;
            a.x = Arow[kk * 4 + kbase + 0];
            a.y = Arow[kk * 4 + kbase + 1];
            // D = A(16x4) * B(4x16) + C   -> v_wmma_f32_16x16x4_f32
            acc = __builtin_amdgcn_wmma_f32_16x16x4_f32(
                /*neg_a=*/false, a, /*neg_b=*/false, bmat,
                /*c_mod=*/(short)0, acc, /*reuse_a=*/false, /*reuse_b=*/false);
        }
        // C/D layout: lanes 0-15 hold M=0..7 in acc[0..7]; lanes 16-31 hold M=8..15.
        // All columns are identical; lane 0 and lane 16 write the 16 row sums.
        if (lane == 0) {
            #pragma unroll
            for (int r = 0; r < 8; ++r) {
                const int d = t * 16 + r;
                c_out[d] = sig * (acc[r] + bv[d]);
            }
        } else if (lane == 16) {
            #pragma unroll
            for (int r = 0; r < 8; ++r) {
                const int d = t * 16 + 8 + r;
                c_out[d] = sig * (acc[r] + bv[d]);
            }
        }
    }
}

// ---- Kernel 2: out = x + c (broadcast over last dim), pure HBM stream ------
__global__ void __launch_bounds__(256)
leea_apply_kernel(const float4* __restrict__ x4,
                  const float*  __restrict__ c,    // [D_MODEL]
                  float4* __restrict__ out4,
                  long long n4)
{
    __shared__ float4 cs[D_MODEL / 4];               // 1 KB constant tile
    if (threadIdx.x < D_MODEL / 4) {
        cs[threadIdx.x] = ((const float4*)c)[threadIdx.x];
    }
    __syncthreads();

    const long long i = (long long)blockIdx.x * blockDim.x + threadIdx.x;
    if (i < n4) {
        // Prefetch a future cacheline of the x stream (speculative, dropped if OOB path).
        const long long pf = i + 32768;
        if (pf < n4) __builtin_prefetch(&x4[pf], 0, 1);

        float4 v  = x4[i];
        // blockDim.x = 256 is a multiple of 64, so (i & 63) == (threadIdx.x & 63)
        float4 cv = cs[threadIdx.x & (D_MODEL / 4 - 1)];
        v.x += cv.x;
        v.y += cv.y;
        v.z += cv.z;
        v.w += cv.w;
        out4[i] = v;
    }
}

// ---- Launcher ---------------------------------------------------------------
extern "C" void kernel_launch(void* const* d_in, const int* in_sizes, int n_in,
                              void* d_out, int out_size, void* d_ws, size_t ws_size,
                              hipStream_t stream) {
    // setup_inputs order: x, mask, distances, Wk, bk, Wv, bv, gate
    const float* x    = (const float*)d_in[0];
    // d_in[1] mask, d_in[2] distances, d_in[3] Wk, d_in[4] bk: provably unused
    const float* Wv   = (const float*)d_in[5];
    const float* bv   = (const float*)d_in[6];
    const float* gate = (const float*)d_in[7];

    float* out = (float*)d_out;
    float* c   = (float*)d_ws;                       // D_MODEL floats of scratch

    leea_const_wmma_kernel<<<1, 32, 0, stream>>>(Wv, bv, gate, c);

    const long long n4 = (long long)out_size / 4;    // 8*2048*256/4 = 1,048,576
    const int threads = 256;
    const int blocks  = (int)((n4 + threads - 1) / threads);
    leea_apply_kernel<<<blocks, threads, 0, stream>>>(
        (const float4*)x, c, (float4*)out, n4);
}